// SubroutineStackFactory_10256381903125
// MI455X (gfx1250) — compile-verified
//
#include <hip/hip_runtime.h>
#include <stdint.h>

// Problem sizes (match reference).
#define DEPTH  8
#define BATCH  8192
#define DMODEL 512
#define LN_EPS 1e-5f

typedef uint32_t u32x4 __attribute__((ext_vector_type(4)));
typedef int      i32x8 __attribute__((ext_vector_type(8)));
typedef int      i32x4 __attribute__((ext_vector_type(4)));

// ---------------------------------------------------------------------------
// CDNA5 async global->LDS copy: GLOBAL_LOAD_ASYNC_TO_LDS_B128 (ASYNCcnt),
// GV addressing mode. LDS destination is the byte offset in the VDST VGPR.
// (Assembled successfully in round 1.)
// ---------------------------------------------------------------------------
__device__ __forceinline__ void async_copy_b128(uint32_t lds_byte_off, const void* gaddr) {
  asm volatile("global_load_async_to_lds_b128 %0, %1, off"
               :: "v"(lds_byte_off), "v"(gaddr)
               : "memory");
}
__device__ __forceinline__ void wait_async_all() {
  asm volatile("s_wait_asynccnt 0x0" ::: "memory");
}

// ---------------------------------------------------------------------------
// CDNA5 Tensor Data Mover: TENSOR_LOAD_TO_LDS with a 2-D tile descriptor.
// Builtin if available (5-arg on ROCm7.2/clang-22, 6-arg on clang>=23),
// inline-asm 2-group form otherwise.
// ---------------------------------------------------------------------------
#if defined(__has_builtin)
#  if __has_builtin(__builtin_amdgcn_tensor_load_to_lds)
#    define HAVE_TDM_BUILTIN 1
#  endif
#endif

__device__ __forceinline__ void tdm_load_2d(u32x4 g0, i32x8 g1) {
#if defined(HAVE_TDM_BUILTIN)
  i32x4 gz = {0, 0, 0, 0};
#  if __clang_major__ >= 23
  i32x8 gz8 = {0, 0, 0, 0, 0, 0, 0, 0};
  __builtin_amdgcn_tensor_load_to_lds(g0, g1, gz, gz, gz8, 0);
#  else
  __builtin_amdgcn_tensor_load_to_lds(g0, g1, gz, gz, 0);
#  endif
#else
  asm volatile("tensor_load_to_lds %0, %1" :: "s"(g0), "s"(g1) : "memory");
#endif
}

__device__ __forceinline__ void wait_tensor_all() {
#if defined(__has_builtin)
#  if __has_builtin(__builtin_amdgcn_s_wait_tensorcnt)
  __builtin_amdgcn_s_wait_tensorcnt((short)0);
  return;
#  endif
#endif
  asm volatile("s_wait_tensorcnt 0x0" ::: "memory");
}

// wave32 all-lanes sum (fixed order -> deterministic).
__device__ __forceinline__ float wave_sum32(float v) {
#pragma unroll
  for (int off = 16; off > 0; off >>= 1) v += __shfl_xor(v, off, 32);
  return v;
}

__global__ __launch_bounds__(256)
void subroutine_stack_kernel(const float* __restrict__ probs,        // (D,B)
                             const float* __restrict__ stack,        // (D,B,M)
                             const float* __restrict__ tensor,       // (B,M)
                             const float* __restrict__ logits,       // (B,3)
                             const float* __restrict__ lnw,          // (M)
                             const float* __restrict__ lnb,          // (M)
                             float* __restrict__ out_read,           // (B,M)
                             float* __restrict__ out_new_probs,      // (D,B)
                             float* __restrict__ out_new_stack,      // (D,B,M)
                             float* __restrict__ out_destack_out,    // (B,M)
                             float* __restrict__ out_destack_prob) { // (B)
  const int b    = blockIdx.x;     // one batch element per block
  const int tid  = threadIdx.x;    // 0..255 (8 wave32s)
  const int wave = tid >> 5;       // wave id == depth handled in stage 2
  const int lane = tid & 31;

  __shared__ float s_stack[DEPTH][DMODEL];  // 16 KB: all depths of row b
  __shared__ float s_tensor[DMODEL];        //  2 KB

  // ---- Stage 0a: TDM DMA of the strided 2-D stack tile into LDS ----
  // Tile: tile_dim0 = 512 elements (one row), tile_dim1 = 8 rows (depths),
  // row stride tensor_dim0_stride = B*M elements (16 MB apart in memory).
  // TDM packs rows contiguously in LDS -> exactly s_stack[d][m] layout.
  if (wave == 0) {
    const uint64_t gaddr = (uint64_t)(uintptr_t)(stack + (size_t)b * DMODEL);
    u32x4 g0 = { 1u,                                        // count=1, user desc
                 (uint32_t)(uintptr_t)(&s_stack[0][0]),     // lds_addr (bytes)
                 (uint32_t)gaddr,                           // global_addr[31:0]
                 (uint32_t)((gaddr >> 32) & 0x01FFFFFFu)    // global_addr[56:32]
                   | (2u << 30) };                          // type=2 ("image")
    i32x8 g1 = { (int)(2u << 16),                   // data_size=4B; no flags
                 (int)((uint32_t)DMODEL << 16),     // tensor_dim0[15:0]=512
                 (int)((uint32_t)DEPTH  << 16),     // dim0 hi=0 | tensor_dim1=8
                 (int)((uint32_t)DMODEL << 16),     // dim1 hi=0 | tile_dim0=512
                 (int)DEPTH,                        // tile_dim1=8 | tile_dim2=0
                 (int)(BATCH * DMODEL),             // tensor_dim0_stride lo
                 0, 0 };                            // stride hi / dim1_stride
    tdm_load_2d(g0, g1);
    wait_tensor_all();   // TENSORcnt is per-wave; only wave 0 issued
  }

  // ---- Stage 0b: async-stage tensor[b, :] into LDS (ASYNCcnt path) ----
  if (tid < DMODEL / 4) {
    const float4* g = (const float4*)(tensor + (size_t)b * DMODEL) + tid;
    async_copy_b128((uint32_t)(uintptr_t)&s_tensor[tid * 4], g);
  }
  wait_async_all();   // each wave drains its own ASYNCcnt (no-op if none)
  __syncthreads();    // publish LDS across the 8 waves

  // ---- Per-b uniform scalars (same for every thread in the block) ----
  float p[DEPTH];
#pragma unroll
  for (int d = 0; d < DEPTH; ++d) p[d] = probs[(size_t)d * BATCH + b];

  const float l0 = logits[(size_t)b * 3 + 0];
  const float l1 = logits[(size_t)b * 3 + 1];
  const float l2 = logits[(size_t)b * 3 + 2];

  // Depths 0..D-2: plain softmax over the 3 logits (destack allowed: 5 >= 1).
  const float mx3 = fmaxf(l0, fmaxf(l1, l2));
  const float e0 = expf(l0 - mx3), e1 = expf(l1 - mx3), e2 = expf(l2 - mx3);
  const float inv3 = 1.0f / (e0 + e1 + e2);
  const float pd_m = e0 * inv3, pn_m = e1 * inv3, pe_m = e2 * inv3;
  // Depth D-1: enstack (action 2) masked to -inf -> softmax over (l0, l1).
  const float mx2 = fmaxf(l0, l1);
  const float f0 = expf(l0 - mx2), f1 = expf(l1 - mx2);
  const float inv2 = 1.0f / (f0 + f1);
  const float pd_l = f0 * inv2, pn_l = f1 * inv2;   // pe_l == 0

  float np[DEPTH], cf[DEPTH];
#pragma unroll
  for (int d = 0; d < DEPTH; ++d) {
    const bool  last = (d == DEPTH - 1);
    const float pd_  = last ? pd_l : pd_m;
    const float pn_  = last ? pn_l : pn_m;
    const float pe_  = last ? 0.0f : pe_m;
    const float en   = (d > 0) ? p[d - 1] : 0.0f;   // en_probs (shift down)
    const float de   = last ? 0.0f : p[d + 1];      // de_probs (shift up)
    np[d] = pe_ * en + pn_ * p[d] + pd_ * de;       // new_probs
    // en_stack == stack, de_stack == stack*(1-p): pre-commit stack = cf * stack
    cf[d] = pe_ + pn_ + pd_ * (1.0f - p[d]);
  }

  // ---- Stage 1: read = sum_d p[d]*stack[d,b,:]; destack outputs ----
  {
    const int m = tid * 2;   // 2 contiguous floats per thread, coalesced
    float r0 = 0.0f, r1 = 0.0f;
#pragma unroll
    for (int d = 0; d < DEPTH; ++d) {
      r0 += p[d] * s_stack[d][m];
      r1 += p[d] * s_stack[d][m + 1];
    }
    const size_t o = (size_t)b * DMODEL + m;
    out_read[o]     = r0;
    out_read[o + 1] = r1;
    out_destack_out[o]     = p[0] * r0;
    out_destack_out[o + 1] = p[0] * r1;
  }
  if (tid < DEPTH) out_new_probs[(size_t)tid * BATCH + b] = np[tid];
  if (tid == 0)    out_destack_prob[b] = p[0];

  // ---- Stage 2: wave d -> fused scale + residual + layernorm + commit ----
  const int   d   = wave;
  const float c   = cf[d];
  const float npv = np[d];

  float4 v[4], t4[4];
  float sum = 0.0f;
#pragma unroll
  for (int k = 0; k < 4; ++k) {
    const int m = lane * 4 + k * 128;                  // 16 floats / lane
    const float4 s4 = *(const float4*)&s_stack[d][m];  // ds_load_b128
    const float4 tt = *(const float4*)&s_tensor[m];
    float4 vv;
    vv.x = fmaf(c, s4.x, tt.x);
    vv.y = fmaf(c, s4.y, tt.y);
    vv.z = fmaf(c, s4.z, tt.z);
    vv.w = fmaf(c, s4.w, tt.w);
    sum += (vv.x + vv.y) + (vv.z + vv.w);
    v[k]  = vv;
    t4[k] = tt;
  }
  const float mu = wave_sum32(sum) * (1.0f / DMODEL);
  float sq = 0.0f;
#pragma unroll
  for (int k = 0; k < 4; ++k) {
    const float dx = v[k].x - mu, dy = v[k].y - mu;
    const float dz = v[k].z - mu, dw = v[k].w - mu;
    sq += (dx * dx + dy * dy) + (dz * dz + dw * dw);
  }
  const float var  = wave_sum32(sq) * (1.0f / DMODEL);
  const float rstd = rsqrtf(var + LN_EPS);
  const float om   = 1.0f - npv;

#pragma unroll
  for (int k = 0; k < 4; ++k) {
    const int m = lane * 4 + k * 128;
    const float4 w4 = *(const float4*)(lnw + m);  // broadcast, L2-resident
    const float4 b4 = *(const float4*)(lnb + m);
    // new_stack = np*LN(v) + (1-np)*(c*stack); note c*stack == v - tensor.
    float4 o;
    o.x = npv * fmaf((v[k].x - mu) * rstd, w4.x, b4.x) + om * (v[k].x - t4[k].x);
    o.y = npv * fmaf((v[k].y - mu) * rstd, w4.y, b4.y) + om * (v[k].y - t4[k].y);
    o.z = npv * fmaf((v[k].z - mu) * rstd, w4.z, b4.z) + om * (v[k].z - t4[k].z);
    o.w = npv * fmaf((v[k].w - mu) * rstd, w4.w, b4.w) + om * (v[k].w - t4[k].w);
    *(float4*)(out_new_stack + ((size_t)d * BATCH + b) * DMODEL + m) = o;
  }
}

extern "C" void kernel_launch(void* const* d_in, const int* in_sizes, int n_in,
                              void* d_out, int out_size, void* d_ws, size_t ws_size,
                              hipStream_t stream) {
  (void)in_sizes; (void)n_in; (void)out_size; (void)d_ws; (void)ws_size;

  const float* probs  = (const float*)d_in[0];  // (8, 8192)
  const float* stack  = (const float*)d_in[1];  // (8, 8192, 512)
  const float* tensor = (const float*)d_in[2];  // (8192, 512)
  const float* logits = (const float*)d_in[3];  // (8192, 3)
  const float* lnw    = (const float*)d_in[4];  // (512)
  const float* lnb    = (const float*)d_in[5];  // (512)

  // Outputs concatenated flat in reference return order.
  float* out      = (float*)d_out;
  float* o_read   = out;
  float* o_nprobs = o_read   + (size_t)BATCH * DMODEL;
  float* o_nstack = o_nprobs + (size_t)DEPTH * BATCH;
  float* o_dout   = o_nstack + (size_t)DEPTH * BATCH * DMODEL;
  float* o_dprob  = o_dout   + (size_t)BATCH * DMODEL;

  dim3 grid(BATCH), block(256);
  hipLaunchKernelGGL(subroutine_stack_kernel, grid, block, 0, stream,
                     probs, stack, tensor, logits, lnw, lnb,
                     o_read, o_nprobs, o_nstack, o_dout, o_dprob);
}